// EquivariantLinear_10342281249296
// MI455X (gfx1250) — compile-verified
//
#include <hip/hip_runtime.h>
#include <cstdint>

// ---------------------------------------------------------------------------
// EquivariantLinear for MI455X (gfx1250, wave32, WMMA).
// out[b,n,d] = sum_m W_{idx(d)}[n,m] * f[b,m,d]  (+ bias[n] on d==0)
// idx = [0, 1,1,1, 2,2,2,2,2];  B=65536, IN=OUT=128, DIM=9, all fp32.
// fp32 matrix path: V_WMMA_F32_16X16X4_F32 (A 16x4, B 4x16, C/D 16x16).
// Roofline: 19.3 GFLOP over ~604 MB HBM traffic -> ~26 us memory floor;
// fp32 WMMA keeps numerics and roughly matches that floor.
// ---------------------------------------------------------------------------

typedef float v2f __attribute__((ext_vector_type(2)));
typedef float v8f __attribute__((ext_vector_type(8)));
typedef int   v4i __attribute__((vector_size(16))); // matches builtin pointee

#define AS_GLOBAL __attribute__((address_space(1)))
#define AS_LDS    __attribute__((address_space(3)))

#ifndef __has_builtin
#define __has_builtin(x) 0
#endif

#if __has_builtin(__builtin_amdgcn_global_load_async_to_lds_b128)
#define USE_ASYNC_COPY 1
#else
#define USE_ASYNC_COPY 0
#endif

namespace {
constexpr int kBatch  = 65536;
constexpr int kIn     = 128;            // IN_MULT
constexpr int kOut    = 128;            // OUT_MULT
constexpr int kDim    = 9;              // 1 + 3 + 5
constexpr int kMT     = 16;             // batch rows per block (WMMA M)
constexpr int kRowF   = kIn * kDim;     // 1152 floats per batch row of f
constexpr int kLdsRow = kRowF + 4;      // 1156: (1156 mod 64)=4 -> conflict-free
constexpr int kThreads = 256;           // 8 wave32
constexpr size_t kLdsBytes = (size_t)kMT * kLdsRow * sizeof(float); // 73984 B
}

// Process one weight matrix G covering irrep dims [D0, D1).
// B-strip for this wave's n-tile is held in 64 VGPRs (v2f bfrag[32]).
// ISA 32-bit layouts (wave32):
//   A 16x4 : lanes 0-15 -> M=lane, V0=K0,V1=K1 ; lanes 16-31 -> K2,K3
//   B 4x16 : lanes 0-15 -> N=lane, V0=K0,V1=K1 ; lanes 16-31 -> K2,K3
//   C 16x16: VGPR r -> M=r (lanes 0-15) / M=r+8 (lanes 16-31), N=lane&15
template <int G, int D0, int D1>
__device__ __forceinline__ void run_group(const float* __restrict__ w,
                                          float* __restrict__ out,
                                          const float* fsrow, // &fs[rowm*kLdsRow]
                                          int b0, int n0, int rowm,
                                          int hi, int klo, float bv) {
  // Preload B strip: rows n0..n0+15 of W_G; this lane covers k = 4j+klo, +1.
  v2f bfrag[32];
  const float* wp = w + (size_t)(G * kOut + n0 + rowm) * kIn + klo;
#pragma unroll
  for (int j = 0; j < 32; ++j)
    bfrag[j] = *(const v2f*)(wp + 4 * j);

#pragma unroll
  for (int d = D0; d < D1; ++d) {
    const float* ap = fsrow + d; // + k*kDim indexes f[b0+rowm][k][d]
    v8f c = {};
#pragma unroll
    for (int j = 0; j < 32; ++j) {
      const int kf = 4 * j + klo;
      v2f a;
      a.x = ap[kf * kDim];
      a.y = ap[(kf + 1) * kDim];
      // D = A(16x4) x B(4x16) + C
      c = __builtin_amdgcn_wmma_f32_16x16x4_f32(
          /*neg_a=*/false, a, /*neg_b=*/false, bfrag[j],
          /*c_mod=*/(short)0, c, /*reuse_a=*/false, /*reuse_b=*/false);
    }
    if (d == 0) { // bias + identity activation on scalar irrep only
#pragma unroll
      for (int r = 0; r < 8; ++r) c[r] += bv;
    }
    // Scatter-store this d's 16x16 tile; stride-36B b32 stores merge in the
    // 192MB write-back L2 into full-line HBM writebacks.
    size_t ob = ((size_t)(b0 + 8 * hi) * kOut + (n0 + rowm)) * kDim + d;
#pragma unroll
    for (int r = 0; r < 8; ++r)
      out[ob + (size_t)r * kOut * kDim] = c[r];
  }
}

__global__ __launch_bounds__(kThreads) void eqlin_wmma_kernel(
    const float* __restrict__ f, const float* __restrict__ w,
    const float* __restrict__ bias, float* __restrict__ out) {
  extern __shared__ float fs[]; // [kMT][kLdsRow] raw copy of f tile, padded

  const int tid = threadIdx.x;
  const int b0  = blockIdx.x * kMT;

  // ---- Stage f[b0..b0+15][0..127][0..8] into LDS: 4608 float4, 18/thread ---
  {
    const float* gsrc = f + (size_t)b0 * kRowF;
#pragma unroll
    for (int j = 0; j < 18; ++j) {
      const int i   = tid + kThreads * j; // float4 index, 0..4607
      const int row = i / 288;            // 288 float4 per batch row
      const int c4  = i - row * 288;
      const float* gp = gsrc + row * kRowF + c4 * 4;
      float*       lp = &fs[row * kLdsRow + c4 * 4];
#if USE_ASYNC_COPY
      __builtin_amdgcn_global_load_async_to_lds_b128(
          (AS_GLOBAL v4i*)gp, (AS_LDS v4i*)lp, 0, 0);
#else
      *(float4*)lp = *(const float4*)gp;
#endif
    }
#if USE_ASYNC_COPY
#if __has_builtin(__builtin_amdgcn_s_wait_asynccnt)
    __builtin_amdgcn_s_wait_asynccnt(0);
#else
    asm volatile("s_wait_asynccnt 0" ::: "memory");
#endif
#endif
    __syncthreads();
  }

  // ---- Per-wave tile: n-tile = 16*waveId, all 9 dims, K = 128 -------------
  const int lane = tid & 31;
  const int wv   = tid >> 5;        // 0..7
  const int n0   = wv * 16;
  const int rowm = lane & 15;       // A: M row; B: N column
  const int hi   = (lane >> 4) & 1; // lane half
  const int klo  = hi * 2;          // K offset of this lane half

  const float bv = bias[n0 + rowm]; // bias is (OUT_MULT, 1) -> flat 128
  const float* fsrow = &fs[rowm * kLdsRow];

  // idx(d): d=0 -> W0 ; d=1..3 -> W1 ; d=4..8 -> W2. Loop over weight matrix
  // so each B-strip is fetched once per block and reused across its dims.
  run_group<0, 0, 1>(w, out, fsrow, b0, n0, rowm, hi, klo, bv);
  run_group<1, 1, 4>(w, out, fsrow, b0, n0, rowm, hi, klo, bv);
  run_group<2, 4, 9>(w, out, fsrow, b0, n0, rowm, hi, klo, bv);
}

extern "C" void kernel_launch(void* const* d_in, const int* in_sizes, int n_in,
                              void* d_out, int out_size, void* d_ws,
                              size_t ws_size, hipStream_t stream) {
  (void)in_sizes; (void)n_in; (void)out_size; (void)d_ws; (void)ws_size;
  const float* f    = (const float*)d_in[0]; // (B, 128, 9)
  const float* w    = (const float*)d_in[1]; // (384, 128)
  const float* bias = (const float*)d_in[2]; // (128, 1)
  // d_in[3] = indices, d_in[4] = scalar_locs: fixed by construction
  // ([0,1,1,1,2,2,2,2,2], [0]) and baked into the kernel's group structure.
  float* out = (float*)d_out;                // (B, 128, 9)

  const dim3 grid(kBatch / kMT); // 4096
  const dim3 block(kThreads);    // 8 wave32
  eqlin_wmma_kernel<<<grid, block, kLdsBytes, stream>>>(f, w, bias, out);
}